// GNN_55576876810521
// MI455X (gfx1250) — compile-verified
//
#include <hip/hip_runtime.h>

#define N_NODES   100000
#define N_EDGES   800000
#define N_ETYPES  5
#define NUM_GRAPHS 64
#define IN_DIM    23
#define IN_PAD    24      // IN_DIM padded to a multiple of 4 (WMMA K-step)
#define HID_DIM   128
#define OUT_DIM   64

typedef float v2f __attribute__((ext_vector_type(2)));
typedef float v8f __attribute__((ext_vector_type(8)));

// ---------------------------------------------------------------- utilities
__global__ void zero_f32(float* __restrict__ p, int n) {
  int i = blockIdx.x * blockDim.x + threadIdx.x;
  if (i < n) p[i] = 0.0f;
}

// degrees for all 5 edge types in one pass (layer-invariant)
__global__ void deg_scatter(const int* __restrict__ edges, float* __restrict__ deg) {
  int e = blockIdx.x * blockDim.x + threadIdx.x;
  int t = blockIdx.y;
  if (e >= N_EDGES) return;
  int d = edges[(t * 2 + 1) * N_EDGES + e];
  atomicAdd(&deg[t * N_NODES + d], 1.0f);
}

// warp-per-edge gather h[src] (stride dim) -> atomic scatter into neigh[dst] (stride ostride)
__global__ void feat_scatter(const float* __restrict__ h,
                             const int* __restrict__ src,
                             const int* __restrict__ dst,
                             float* __restrict__ neigh, int dim, int ostride) {
  int warp = (blockIdx.x * blockDim.x + threadIdx.x) >> 5;
  int lane = threadIdx.x & 31;
  if (warp >= N_EDGES) return;
  int s = src[warp];
  int d = dst[warp];
  const float* hp = h + (size_t)s * dim;
  float*       np = neigh + (size_t)d * ostride;
  for (int j = lane; j < dim; j += 32)
    atomicAdd(&np[j], hp[j]);
}

// In-place: A[n, j] = (A[n, j] + h[n, j]) / (deg[n] + 1); padded col (j >= DIN) stays 0.
// Branchless: clamped load + value select, no exec-mask churn.
template<int DIN, int KP>
__global__ void prep_a(float* __restrict__ a, const float* __restrict__ h,
                       const float* __restrict__ deg) {
  int i = blockIdx.x * blockDim.x + threadIdx.x;
  if (i >= N_NODES * KP) return;
  int n = i / KP;
  int j = i - n * KP;
  float invd = 1.0f / (deg[n] + 1.0f);
  int jc = (j < DIN) ? j : (DIN - 1);
  float hv = h[(size_t)n * DIN + jc];
  hv = (j < DIN) ? hv : 0.0f;
  a[i] = (a[i] + hv) * invd;
}

// Zero-pad W1_t [23,128] -> Wp [24,128] (same linear index for valid rows)
__global__ void pad_w1(const float* __restrict__ W, float* __restrict__ Wp) {
  int i = blockIdx.x * blockDim.x + threadIdx.x;
  if (i >= IN_PAD * HID_DIM) return;
  int row = i / HID_DIM;
  Wp[i] = (row < IN_DIM) ? W[i] : 0.0f;
}

// out[n, j] = sum_t b[t, j]   (seed accumulator with summed bias)
__global__ void bias_init(const float* __restrict__ b, float* __restrict__ out, int dim) {
  int i = blockIdx.x * blockDim.x + threadIdx.x;
  if (i >= N_NODES * dim) return;
  int j = i % dim;
  float s = 0.0f;
#pragma unroll
  for (int t = 0; t < N_ETYPES; ++t) s += b[t * dim + j];
  out[i] = s;
}

// ------------------------------------------------------ WMMA f32 GEMM tile
// out[16x16 tile] += A @ W with A pre-formed/padded -> branchless unrolled K loop.
// A fragment (16x4 f32): lanes 0-15 hold K=k0,k0+1 of row M=lane (one b64 load),
// lanes 16-31 hold K=k0+2,k0+3. One wave per tile; threadIdx.y = N-tile.
template<int KP, int DOUT>
__global__ void sage_gemm_wmma(const float* __restrict__ A,   // [N_NODES, KP]
                               const float* __restrict__ W,   // [KP, DOUT]
                               float* __restrict__ out) {     // [N_NODES, DOUT] accum
  const int lane  = threadIdx.x;              // 0..31
  const int mbase = blockIdx.x * 16;
  const int nbase = threadIdx.y * 16;
  const int r16   = lane & 15;
  const int half  = lane >> 4;                // K-pair selector
  const int m     = mbase + r16;
  const int n     = nbase + r16;

  // load C tile (accumulate across etypes; bias pre-seeded)
  v8f c;
  const int crow = mbase + half * 8;          // M = crow + r
#pragma unroll
  for (int r = 0; r < 8; ++r)
    c[r] = out[(size_t)(crow + r) * DOUT + n];

  const float* arow = A + (size_t)m * KP + half * 2;   // 8B-aligned (KP even)
  const float* wcol = W + (size_t)(half * 2) * DOUT + n;

#pragma unroll
  for (int k0 = 0; k0 < KP; k0 += 4) {
    v2f a = *(const v2f*)(arow + k0);
    v2f b;
    b.x = wcol[(size_t)k0 * DOUT];
    b.y = wcol[(size_t)k0 * DOUT + DOUT];
    c = __builtin_amdgcn_wmma_f32_16x16x4_f32(
        /*neg_a=*/false, a, /*neg_b=*/false, b,
        /*c_mod=*/(short)0, c, /*reuse_a=*/false, /*reuse_b=*/false);
  }

#pragma unroll
  for (int r = 0; r < 8; ++r)
    out[(size_t)(crow + r) * DOUT + n] = c[r];
}

__global__ void relu_inplace(float* __restrict__ p, int n) {
  int i = blockIdx.x * blockDim.x + threadIdx.x;
  if (i < n) p[i] = fmaxf(p[i], 0.0f);
}

// warp-per-node mean-pool accumulate
__global__ void pool_scatter(const float* __restrict__ h2,
                             const int* __restrict__ gids,
                             float* __restrict__ pool, float* __restrict__ cnt) {
  int node = (blockIdx.x * blockDim.x + threadIdx.x) >> 5;
  int lane = threadIdx.x & 31;
  if (node >= N_NODES) return;
  int g = gids[node];
  const float* hp = h2 + (size_t)node * OUT_DIM;
#pragma unroll
  for (int j = 0; j < OUT_DIM; j += 32)
    atomicAdd(&pool[g * OUT_DIM + j + lane], hp[j + lane]);
  if (lane == 0) atomicAdd(&cnt[g], 1.0f);
}

__global__ void pool_divide(float* __restrict__ pool, const float* __restrict__ cnt) {
  int i = blockIdx.x * blockDim.x + threadIdx.x;
  if (i >= NUM_GRAPHS * OUT_DIM) return;
  pool[i] = pool[i] / fmaxf(cnt[i / OUT_DIM], 1.0f);
}

// ---------------------------------------------------------------- launcher
extern "C" void kernel_launch(void* const* d_in, const int* in_sizes, int n_in,
                              void* d_out, int out_size, void* d_ws, size_t ws_size,
                              hipStream_t stream) {
  const float* x     = (const float*)d_in[0];
  const int*   edges = (const int*)  d_in[1];
  const int*   gids  = (const int*)  d_in[2];
  const float* W1    = (const float*)d_in[3];
  const float* b1    = (const float*)d_in[4];
  const float* W2    = (const float*)d_in[5];
  const float* b2    = (const float*)d_in[6];
  float* out = (float*)d_out;

  float* ws    = (float*)d_ws;
  float* deg   = ws;                                   // 5 * N
  float* neigh = deg   + (size_t)N_ETYPES * N_NODES;   // N * 128 (also holds A)
  float* h1    = neigh + (size_t)N_NODES * HID_DIM;    // N * 128
  float* h2    = h1    + (size_t)N_NODES * HID_DIM;    // N * 64
  float* cnt   = h2    + (size_t)N_NODES * OUT_DIM;    // 64
  float* Wpad  = cnt   + NUM_GRAPHS;                   // 24 * 128

  const int T256 = 256;
  auto blks = [](int n, int t) { return (n + t - 1) / t; };

  // ---- degrees (shared by both layers) ----
  zero_f32<<<blks(N_ETYPES * N_NODES, T256), T256, 0, stream>>>(deg, N_ETYPES * N_NODES);
  deg_scatter<<<dim3(blks(N_EDGES, T256), N_ETYPES), T256, 0, stream>>>(edges, deg);

  // ---- layer 1: x[N,23] -> h1[N,128] ----
  bias_init<<<blks(N_NODES * HID_DIM, T256), T256, 0, stream>>>(b1, h1, HID_DIM);
  for (int t = 0; t < N_ETYPES; ++t) {
    zero_f32<<<blks(N_NODES * IN_PAD, T256), T256, 0, stream>>>(neigh, N_NODES * IN_PAD);
    feat_scatter<<<blks(N_EDGES * 32, T256), T256, 0, stream>>>(
        x, edges + (size_t)(t * 2 + 0) * N_EDGES,
           edges + (size_t)(t * 2 + 1) * N_EDGES, neigh, IN_DIM, IN_PAD);
    prep_a<IN_DIM, IN_PAD><<<blks(N_NODES * IN_PAD, T256), T256, 0, stream>>>(
        neigh, x, deg + (size_t)t * N_NODES);
    pad_w1<<<blks(IN_PAD * HID_DIM, T256), T256, 0, stream>>>(
        W1 + (size_t)t * IN_DIM * HID_DIM, Wpad);
    sage_gemm_wmma<IN_PAD, HID_DIM>
        <<<dim3(N_NODES / 16, 1), dim3(32, HID_DIM / 16), 0, stream>>>(neigh, Wpad, h1);
  }
  relu_inplace<<<blks(N_NODES * HID_DIM, T256), T256, 0, stream>>>(h1, N_NODES * HID_DIM);

  // ---- layer 2: h1[N,128] -> h2[N,64] ----
  bias_init<<<blks(N_NODES * OUT_DIM, T256), T256, 0, stream>>>(b2, h2, OUT_DIM);
  for (int t = 0; t < N_ETYPES; ++t) {
    zero_f32<<<blks(N_NODES * HID_DIM, T256), T256, 0, stream>>>(neigh, N_NODES * HID_DIM);
    feat_scatter<<<blks(N_EDGES * 32, T256), T256, 0, stream>>>(
        h1, edges + (size_t)(t * 2 + 0) * N_EDGES,
            edges + (size_t)(t * 2 + 1) * N_EDGES, neigh, HID_DIM, HID_DIM);
    prep_a<HID_DIM, HID_DIM><<<blks(N_NODES * HID_DIM, T256), T256, 0, stream>>>(
        neigh, h1, deg + (size_t)t * N_NODES);
    sage_gemm_wmma<HID_DIM, OUT_DIM>
        <<<dim3(N_NODES / 16, 1), dim3(32, OUT_DIM / 16), 0, stream>>>(
        neigh, W2 + (size_t)t * HID_DIM * OUT_DIM, h2);
  }

  // ---- mean pool over graphs ----
  zero_f32<<<blks(NUM_GRAPHS * OUT_DIM, T256), T256, 0, stream>>>(out, NUM_GRAPHS * OUT_DIM);
  zero_f32<<<1, 64, 0, stream>>>(cnt, NUM_GRAPHS);
  pool_scatter<<<blks(N_NODES * 32, T256), T256, 0, stream>>>(h2, gids, out, cnt);
  pool_divide<<<blks(NUM_GRAPHS * OUT_DIM, T256), T256, 0, stream>>>(out, cnt);
}